// ProteinGraphConv_73658689126889
// MI455X (gfx1250) — compile-verified
//
#include <hip/hip_runtime.h>

typedef __attribute__((ext_vector_type(2))) float v2f;
typedef __attribute__((ext_vector_type(8))) float v8f;

#define N_NODES 10000
#define D 128
#define N_TILES 625  // 10000 / 16, exact

// XOR bank swizzle for transposed W in LDS: ldsW[swz(k,n)] holds W[k][n].
// Keeps (k,k+1) pairs contiguous (k even) so B fragments load as ds_load_b64,
// while spreading the 32 lanes of a B-fragment read across all 64 LDS banks.
__device__ __forceinline__ int swz(int k, int n) {
  int kx = k ^ ((n & 31) << 1) ^ ((k & 2) << 4);
  return n * D + kx;
}

// ---------------------------------------------------------------------------
// Kernel 1: h = x @ W + b via v_wmma_f32_16x16x4_f32; also writes agg = h.
// One wave (32 lanes) computes a 16x128 output slab: 8 accumulators of v8f,
// 8 WMMA per K-step, 32 K-steps. W (64KB) is staged transposed+swizzled into
// LDS by the gfx1250 async-to-LDS engine (ASYNCcnt DMA): each lane's per-lane
// LDS destination VGPR performs the transpose during the copy, so no VGPR
// round-trip and no ds_store scatter are needed.
// ---------------------------------------------------------------------------
__global__ __launch_bounds__(256) void gemm_wmma_f32(
    const float* __restrict__ x, const float* __restrict__ W,
    const float* __restrict__ bias, float* __restrict__ h,
    float* __restrict__ agg) {
  __shared__ float ldsW[D * D];  // 64 KB

  const int t = threadIdx.x;

  // Async DMA stage: 16384 dwords over 256 threads = 64 b32 per thread.
  {
    const unsigned ldsBase = (unsigned)(uintptr_t)(&ldsW[0]);  // low32 = LDS offset
    for (int i = 0; i < 64; ++i) {
      const int idx = t + i * 256;  // element index into row-major W
      const int r = idx >> 7;       // K
      const int c = idx & 127;      // N
      const unsigned dst = ldsBase + 4u * (unsigned)swz(r, c);
      const unsigned long long src = (unsigned long long)(uintptr_t)(W + idx);
      asm volatile("global_load_async_to_lds_b32 %0, %1, off"
                   :
                   : "v"(dst), "v"(src)
                   : "memory");
    }
    asm volatile("s_wait_asynccnt 0x0" ::: "memory");
  }
  __syncthreads();

  const int wave = t >> 5;
  const int lane = t & 31;
  const int half = lane >> 4;  // selects K-pair within a 16x4 A fragment
  const int lr   = lane & 15;
  const int tile = blockIdx.x * 8 + wave;
  if (tile >= N_TILES) return;
  const int m0 = tile * 16;

  v8f acc[8] = {};  // 8 x (16x16 f32 accumulator)

  const float* xrow = x + (size_t)(m0 + lr) * D;

  for (int k = 0; k < D; k += 4) {
    const int kk = k + 2 * half;
    // A fragment 16x4: lanes 0-15 hold K={k,k+1}, lanes 16-31 K={k+2,k+3}.
    const float2 av = *(const float2*)(xrow + kk);
    v2f a; a.x = av.x; a.y = av.y;
#pragma unroll
    for (int nt = 0; nt < 8; ++nt) {
      const int n = nt * 16 + lr;
      // B fragment 4x16 from swizzled LDS (contiguous pair -> ds_load_b64).
      const float2 bv = *(const float2*)(&ldsW[swz(kk, n)]);
      v2f b; b.x = bv.x; b.y = bv.y;
      acc[nt] = __builtin_amdgcn_wmma_f32_16x16x4_f32(
          false, a, false, b, (short)0, acc[nt], false, false);
    }
  }

  // D layout: VGPR j -> row m0 + j + 8*half, col n (lane lr).
#pragma unroll
  for (int nt = 0; nt < 8; ++nt) {
    const int n = nt * 16 + lr;
    const float bv = bias[n];
#pragma unroll
    for (int j = 0; j < 8; ++j) {
      const size_t off = (size_t)(m0 + j + 8 * half) * D + n;
      const float v = acc[nt][j] + bv;
      h[off] = v;
      agg[off] = v;  // agg starts as h (self term)
    }
  }
}

// ---------------------------------------------------------------------------
// Kernel 2: per-edge scatter  agg[dst][:] += h[src][:]
// One wave per edge: lane reads float4 of h[src] (coalesced 512B gather, L2-
// resident since h is 5.12MB << 192MB L2), then 4 native global_atomic_add_f32
// (non-returning, L2-side RMW) into agg[dst]. This phase is L2-atomic bound,
// not HBM bound: 327MB of gathers + 82M f32 atomics, all L2-resident.
// ---------------------------------------------------------------------------
__global__ __launch_bounds__(256) void scatter_edges(
    const int* __restrict__ ei, const float* __restrict__ h,
    float* __restrict__ agg, int E) {
  const int lane = threadIdx.x & 31;
  const int edge = blockIdx.x * 8 + (threadIdx.x >> 5);
  if (edge >= E) return;
  const int src = ei[edge];
  const int dst = ei[E + edge];
  const float4 v = ((const float4*)(h + (size_t)src * D))[lane];
  float* base = agg + (size_t)dst * D + lane * 4;
  unsafeAtomicAdd(base + 0, v.x);
  unsafeAtomicAdd(base + 1, v.y);
  unsafeAtomicAdd(base + 2, v.z);
  unsafeAtomicAdd(base + 3, v.w);
}

// ---------------------------------------------------------------------------
// Kernel 3: LayerNorm + ReLU, one wave per row (wave32: 5-step shfl_xor tree).
// ---------------------------------------------------------------------------
__global__ __launch_bounds__(256) void ln_relu(
    const float* __restrict__ agg, const float* __restrict__ lnw,
    const float* __restrict__ lnb, float* __restrict__ out) {
  const int lane = threadIdx.x & 31;
  const int row = blockIdx.x * 8 + (threadIdx.x >> 5);
  if (row >= N_NODES) return;

  const float4 v = ((const float4*)(agg + (size_t)row * D))[lane];
  float s  = v.x + v.y + v.z + v.w;
  float sq = v.x * v.x + v.y * v.y + v.z * v.z + v.w * v.w;
#pragma unroll
  for (int m = 16; m >= 1; m >>= 1) {
    s  += __shfl_xor(s, m, 32);
    sq += __shfl_xor(sq, m, 32);
  }
  const float mean = s * (1.0f / 128.0f);
  const float var  = sq * (1.0f / 128.0f) - mean * mean;
  const float rs   = rsqrtf(var + 1e-5f);

  const float4 w4 = ((const float4*)lnw)[lane];
  const float4 b4 = ((const float4*)lnb)[lane];
  float4 o;
  o.x = fmaxf(0.0f, (v.x - mean) * rs * w4.x + b4.x);
  o.y = fmaxf(0.0f, (v.y - mean) * rs * w4.y + b4.y);
  o.z = fmaxf(0.0f, (v.z - mean) * rs * w4.z + b4.z);
  o.w = fmaxf(0.0f, (v.w - mean) * rs * w4.w + b4.w);
  ((float4*)(out + (size_t)row * D))[lane] = o;
}

extern "C" void kernel_launch(void* const* d_in, const int* in_sizes, int n_in,
                              void* d_out, int out_size, void* d_ws,
                              size_t ws_size, hipStream_t stream) {
  const float* x   = (const float*)d_in[0];
  const int*   ei  = (const int*)d_in[1];   // edge_index, [2, E] flat
  const float* fcw = (const float*)d_in[2];
  const float* fcb = (const float*)d_in[3];
  const float* lnw = (const float*)d_in[4];
  const float* lnb = (const float*)d_in[5];
  float* out = (float*)d_out;

  const int E = in_sizes[1] / 2;

  float* h   = (float*)d_ws;                       // 10000*128 f32 = 5.12 MB
  float* agg = h + (size_t)N_NODES * D;            // another 5.12 MB

  // 625 row-tiles of 16, 8 waves per block -> 79 blocks.
  gemm_wmma_f32<<<(N_TILES + 7) / 8, 256, 0, stream>>>(x, fcw, fcb, h, agg);
  // One wave per edge, 8 edges per block.
  scatter_edges<<<(E + 7) / 8, 256, 0, stream>>>(ei, h, agg, E);
  // One wave per row.
  ln_relu<<<(N_NODES + 7) / 8, 256, 0, stream>>>(agg, lnw, lnb, out);
}